// IntervalDetectionLoss_7258494730638
// MI455X (gfx1250) — compile-verified
//
#include <hip/hip_runtime.h>
#include <hip/hip_bf16.h>

#define NB 8
#define NA 100000
#define NG 64
#define NACT 30
#define NAG 4
#define POS_T_C 0.5f
#define ALPHA_C 0.25f
#define EPS_C 1e-8f

typedef float v2f __attribute__((ext_vector_type(2)));
typedef float v8f __attribute__((ext_vector_type(8)));

__device__ __forceinline__ uint32_t lds_byte_addr(const void* p) {
    // Flat shared-aperture addresses carry the workgroup-relative LDS byte
    // offset in the low 32 bits (ISA §10.2 aperture calc).
    return (uint32_t)(uintptr_t)p;
}

__global__ __launch_bounds__(256) void idl_main_kernel(
    const float* __restrict__ action_logits,
    const float* __restrict__ agent_logits,
    const float* __restrict__ target_logits,
    const float* __restrict__ boundary_offsets,
    const float* __restrict__ objectness,
    const float* __restrict__ anchors,
    const float* __restrict__ gt_boxes,
    const int* __restrict__ gt_action,
    const int* __restrict__ gt_agent,
    const int* __restrict__ gt_target,
    float* __restrict__ partials)
{
    __shared__ __align__(16) float s_box[2 * NG];   // interleaved {start,end}
    __shared__ __align__(16) int   s_act[NG];
    __shared__ __align__(16) int   s_agt[NG];
    __shared__ __align__(16) int   s_tgt[NG];
    __shared__ float sred[8][8];                    // [wave][quantity]

    const int b   = blockIdx.y;
    const int tid = threadIdx.x;
    const int a   = blockIdx.x * 256 + tid;
    const bool valid = (a < NA);
    const int ac  = valid ? a : (NA - 1);

    // ---- Async DMA the per-batch GT data into LDS (ASYNCcnt path).
    // GVS addressing: mem = SADDR(64b sgpr) + VADDR(32b voffset); VDST holds
    // the LDS byte address. 64 lanes move 1.25 KB in 4 async issues while the
    // rest of the block proceeds to its anchor/objectness loads.
    if (tid < NG) {
        const float* gbox = gt_boxes  + (size_t)b * NG * 2;
        const int*   gact = gt_action + (size_t)b * NG;
        const int*   gagt = gt_agent  + (size_t)b * NG;
        const int*   gtgt = gt_target + (size_t)b * NG;
        uint32_t vbox = (uint32_t)tid * 8u;
        uint32_t vint = (uint32_t)tid * 4u;
        uint32_t lbox = lds_byte_addr(&s_box[2 * tid]);
        uint32_t lact = lds_byte_addr(&s_act[tid]);
        uint32_t lagt = lds_byte_addr(&s_agt[tid]);
        uint32_t ltgt = lds_byte_addr(&s_tgt[tid]);
        asm volatile("global_load_async_to_lds_b64 %0, %1, %2"
                     :: "v"(lbox), "v"(vbox), "s"(gbox) : "memory");
        asm volatile("global_load_async_to_lds_b32 %0, %1, %2"
                     :: "v"(lact), "v"(vint), "s"(gact) : "memory");
        asm volatile("global_load_async_to_lds_b32 %0, %1, %2"
                     :: "v"(lagt), "v"(vint), "s"(gagt) : "memory");
        asm volatile("global_load_async_to_lds_b32 %0, %1, %2"
                     :: "v"(ltgt), "v"(vint), "s"(gtgt) : "memory");
    }

    // Anchor geometry + objectness loads overlap the async DMA.
    const float a_c = anchors[(size_t)ac * 2 + 0];
    const float a_s = anchors[(size_t)ac * 2 + 1];
    const float a_start = a_c - 0.5f * a_s;
    const float a_end   = a_c + 0.5f * a_s;
    const float a_len   = a_end - a_start;
    const float obj_x = objectness[(size_t)b * NA + ac];

#if __has_builtin(__builtin_amdgcn_s_wait_asynccnt)
    __builtin_amdgcn_s_wait_asynccnt(0);
#else
    asm volatile("s_wait_asynccnt 0" ::: "memory");
#endif
    __syncthreads();

    // Best-IoU argmax over 64 GT boxes (first-max semantics like jnp.argmax)
    float best = -1.0f;
    int bg = 0;
    #pragma unroll 4
    for (int g = 0; g < NG; ++g) {
        const float2 gb = *(const float2*)&s_box[2 * g];   // ds_load_b64
        float inter = fminf(a_end, gb.y) - fmaxf(a_start, gb.x);
        inter = fmaxf(inter, 0.0f);
        const float uni = a_len + (gb.y - gb.x) - inter;
        const float iou = (inter > 0.0f && uni > 0.0f) ? (inter / uni) : 0.0f;
        if (iou > best) { best = iou; bg = g; }
    }
    const bool pos = valid && (best >= POS_T_C);
    const float posf = pos ? 1.0f : 0.0f;

    const size_t ba = (size_t)b * NA + ac;
    const float* __restrict__ al = action_logits + ba * NACT;
    const float* __restrict__ gl = agent_logits  + ba * NAG;
    const float* __restrict__ tl = target_logits + ba * NAG;
    const float* __restrict__ bo = boundary_offsets + ba * 2;

    if (pos) {
        // gfx1250 global_prefetch_b8: pull the pos-gated rows while we do the
        // focal-loss transcendentals below.
        __builtin_prefetch(al, 0, 0);
        __builtin_prefetch(gl, 0, 0);
        __builtin_prefetch(tl, 0, 0);
        __builtin_prefetch(bo, 0, 0);
    }

    // Focal objectness loss (every anchor; obj_loss is a mean over B*A)
    float obj_term = 0.0f;
    {
        const float p  = 1.0f / (1.0f + expf(-obj_x));
        const float pt = posf * p + (1.0f - posf) * (1.0f - p);
        const float aw = posf * ALPHA_C + (1.0f - posf) * (1.0f - ALPHA_C);
        const float lg = posf * logf(p + EPS_C) + (1.0f - posf) * logf(1.0f - p + EPS_C);
        const float omt = 1.0f - pt;
        obj_term = valid ? (-aw * omt * omt * lg) : 0.0f;
    }

    // Pos-gated terms: CE over 30/4/4 logits + boundary 1-IoU term.
    float act_ce = 0.0f, agt_ce = 0.0f, tgt_ce = 0.0f, bnd = 0.0f;
    if (pos) {
        const int aid  = s_act[bg];
        const int gid  = s_agt[bg] - 1;
        const int tidx = s_tgt[bg] - 1;

        // action CE (log-softmax over 30)
        {
            float m = -3.4e38f;
            float l[NACT];
            #pragma unroll
            for (int i = 0; i < NACT; ++i) { l[i] = al[i]; m = fmaxf(m, l[i]); }
            float s = 0.0f;
            #pragma unroll
            for (int i = 0; i < NACT; ++i) s += expf(l[i] - m);
            act_ce = (logf(s) + m) - l[aid];
        }
        // agent CE (4)
        {
            float l0 = gl[0], l1 = gl[1], l2 = gl[2], l3 = gl[3];
            float m = fmaxf(fmaxf(l0, l1), fmaxf(l2, l3));
            float s = expf(l0 - m) + expf(l1 - m) + expf(l2 - m) + expf(l3 - m);
            float lsel = (gid == 0) ? l0 : (gid == 1) ? l1 : (gid == 2) ? l2 : l3;
            agt_ce = (logf(s) + m) - lsel;
        }
        // target CE (4)
        {
            float l0 = tl[0], l1 = tl[1], l2 = tl[2], l3 = tl[3];
            float m = fmaxf(fmaxf(l0, l1), fmaxf(l2, l3));
            float s = expf(l0 - m) + expf(l1 - m) + expf(l2 - m) + expf(l3 - m);
            float lsel = (tidx == 0) ? l0 : (tidx == 1) ? l1 : (tidx == 2) ? l2 : l3;
            tgt_ce = (logf(s) + m) - lsel;
        }
        // boundary 1-IoU (replicate reference's encode/decode round-trip)
        {
            const float gs_m = s_box[2 * bg + 0], ge_m = s_box[2 * bg + 1];
            const float bt_s = (gs_m - a_start) / a_s;
            const float bt_e = (ge_m - a_end) / a_s;
            const float o0 = bo[0], o1 = bo[1];
            const float ps  = a_start + o0 * a_s;
            const float pe  = a_end   + o1 * a_s;
            const float gs2 = a_start + bt_s * a_s;
            const float ge2 = a_end   + bt_e * a_s;
            float inter2 = fminf(pe, ge2) - fmaxf(ps, gs2);
            inter2 = fmaxf(inter2, 0.0f);
            const float uni2 = (pe - ps) + (ge2 - gs2) - inter2;
            bnd = 1.0f - inter2 / (uni2 + EPS_C);
        }
    }

    // ---- Reduction: 6 quantities, wave32 butterfly then cross-wave via WMMA.
    float v[6] = { obj_term, posf, act_ce, agt_ce, tgt_ce, bnd };
    #pragma unroll
    for (int q = 0; q < 6; ++q) {
        #pragma unroll
        for (int off = 16; off >= 1; off >>= 1)
            v[q] += __shfl_xor(v[q], off, 32);
    }

    const int wave = tid >> 5;
    const int lane = tid & 31;
    if (lane == 0) {
        #pragma unroll
        for (int q = 0; q < 6; ++q) sred[wave][q] = v[q];
        sred[wave][6] = 0.0f;
        sred[wave][7] = 0.0f;
    }
    __syncthreads();

    if (wave == 0) {
        // A[m][k] = sred[k][m&7]; B = ones => D[m][n] = sum_w sred[w][m].
        // ISA A-layout (32-bit 16x4): lanes 0-15 carry K0 (vgpr0) / K1 (vgpr1),
        // lanes 16-31 carry K2 / K3, row M = lane & 15.
        const int q = lane & 7;          // row -> quantity (rows 6..15 don't-care)
        const bool hi = (lane >= 16);
        v2f a01, a23, bones;
        a01[0] = hi ? sred[2][q] : sred[0][q];
        a01[1] = hi ? sred[3][q] : sred[1][q];
        a23[0] = hi ? sred[6][q] : sred[4][q];
        a23[1] = hi ? sred[7][q] : sred[5][q];
        bones[0] = 1.0f; bones[1] = 1.0f;

        v8f acc = {};
        acc = __builtin_amdgcn_wmma_f32_16x16x4_f32(false, a01, false, bones,
                                                    (short)0, acc, false, false);
        acc = __builtin_amdgcn_wmma_f32_16x16x4_f32(false, a23, false, bones,
                                                    (short)0, acc, false, false);
        // D row m, col 0 lives in acc[m] of lane 0.
        if (lane == 0) {
            float* P = partials + ((size_t)blockIdx.y * gridDim.x + blockIdx.x) * 8;
            #pragma unroll
            for (int qq = 0; qq < 6; ++qq) P[qq] = acc[qq];
        }
    }
}

__global__ __launch_bounds__(256) void idl_final_kernel(
    const float* __restrict__ partials, int nblk, float* __restrict__ out)
{
    __shared__ float sred[8][8];
    const int tid = threadIdx.x;
    float acc[6] = {0.f, 0.f, 0.f, 0.f, 0.f, 0.f};
    for (int i = tid; i < nblk; i += 256) {
        const float* P = partials + (size_t)i * 8;
        #pragma unroll
        for (int q = 0; q < 6; ++q) acc[q] += P[q];
    }
    #pragma unroll
    for (int q = 0; q < 6; ++q) {
        #pragma unroll
        for (int off = 16; off >= 1; off >>= 1)
            acc[q] += __shfl_xor(acc[q], off, 32);
    }
    const int wave = tid >> 5;
    const int lane = tid & 31;
    if (lane == 0) {
        #pragma unroll
        for (int q = 0; q < 6; ++q) sred[wave][q] = acc[q];
    }
    __syncthreads();
    if (tid == 0) {
        float s[6];
        #pragma unroll
        for (int q = 0; q < 6; ++q) {
            float t = 0.0f;
            #pragma unroll
            for (int w = 0; w < 8; ++w) t += sred[w][q];
            s[q] = t;
        }
        const float obj    = s[0] / (float)(NB * NA);
        const float npos   = fmaxf(s[1], 1.0f);
        const float action = s[2] / npos;
        const float agent  = s[3] / npos;
        const float target = s[4] / npos;
        const float bnd    = s[5] / npos;
        out[0] = obj + action + agent + target + bnd;
        out[1] = obj;
        out[2] = action;
        out[3] = agent;
        out[4] = target;
        out[5] = bnd;
    }
}

extern "C" void kernel_launch(void* const* d_in, const int* in_sizes, int n_in,
                              void* d_out, int out_size, void* d_ws, size_t ws_size,
                              hipStream_t stream) {
    const float* action_logits    = (const float*)d_in[0];
    const float* agent_logits     = (const float*)d_in[1];
    const float* target_logits    = (const float*)d_in[2];
    const float* boundary_offsets = (const float*)d_in[3];
    const float* objectness       = (const float*)d_in[4];
    const float* anchors          = (const float*)d_in[5];
    const float* gt_boxes         = (const float*)d_in[6];
    const int*   gt_action        = (const int*)d_in[7];
    const int*   gt_agent         = (const int*)d_in[8];
    const int*   gt_target        = (const int*)d_in[9];

    float* partials = (float*)d_ws;
    const int nbx = (NA + 255) / 256;       // 391
    dim3 grid(nbx, NB);                     // 3128 blocks of 8 waves
    idl_main_kernel<<<grid, 256, 0, stream>>>(
        action_logits, agent_logits, target_logits, boundary_offsets,
        objectness, anchors, gt_boxes, gt_action, gt_agent, gt_target, partials);

    const int nblk = nbx * NB;
    idl_final_kernel<<<1, 256, 0, stream>>>(partials, nblk, (float*)d_out);
}